// DiceEmbedding_83562883711776
// MI455X (gfx1250) — compile-verified
//
#include <hip/hip_runtime.h>
#include <math.h>

#define D_POLAR 10
#define KPAD    12          // K padded to multiple of 4 for wmma f32 16x16x4
#define EMB     1024

typedef __attribute__((ext_vector_type(2))) float v2f;
typedef __attribute__((ext_vector_type(8))) float v8f;

// ---------------------------------------------------------------------------
// Kernel 1: fold Q into W.  Bmat[k][e] = (W @ Q)[e,k]  for k<10, zero-padded
// to k=10,11.  Bmat is laid out row-major [KPAD][EMB] so the WMMA B-fragment
// loads are simple column reads.
// ---------------------------------------------------------------------------
__global__ __launch_bounds__(256) void build_B_kernel(
    const float* __restrict__ W,   // [EMB, 10] row-major
    const float* __restrict__ Q,   // [10, 10]  row-major
    float* __restrict__ Bmat)      // [KPAD, EMB]
{
    int e = blockIdx.x * blockDim.x + threadIdx.x;
    if (e >= EMB) return;
    float w[D_POLAR];
#pragma unroll
    for (int j = 0; j < D_POLAR; ++j) w[j] = W[e * D_POLAR + j];
#pragma unroll
    for (int d = 0; d < D_POLAR; ++d) {
        float acc = 0.f;
#pragma unroll
        for (int k = 0; k < D_POLAR; ++k) acc += w[k] * Q[k * D_POLAR + d];
        Bmat[d * EMB + e] = acc;
    }
    Bmat[10 * EMB + e] = 0.f;
    Bmat[11 * EMB + e] = 0.f;
}

// ---------------------------------------------------------------------------
// Kernel 2: per wave: build polar rows for 16 inputs, then sweep 64 column
// tiles with V_WMMA_F32_16X16X4_F32 (3 per tile, K = 12).
// ---------------------------------------------------------------------------
__global__ __launch_bounds__(256) void dice_embed_kernel(
    const float* __restrict__ vals,   // [N]
    const float* __restrict__ Bmat,   // [KPAD, EMB]
    const float* __restrict__ bias,   // [EMB]
    float* __restrict__ out,          // [N, EMB]
    int n)
{
    // 8 waves/block, each wave owns a 16-row n-tile; stride-12 rows are
    // bank-conflict-free (gcd(12,64)=4, 12*m mod 64 distinct for m<16).
    __shared__ float polar[8][16][KPAD];

    const int tid  = (int)threadIdx.x;
    const int wave = tid >> 5;
    const int lane = tid & 31;
    const int nBase = ((int)blockIdx.x * 8 + wave) * 16;
    if (nBase + 16 > n) return;   // wave-uniform: EXEC stays all-ones below

    // --- build polar rows: lanes 0..15, one row each -----------------------
    if (lane < 16) {
        float v     = vals[nBase + lane];
        float theta = logf(0.01f + fabsf(v)) * (float)(M_PI / 85.0);
        float s = sinf(theta);
        float c = cosf(theta);
        float* row = polar[wave][lane];
        // [c, s*c, s^2*c, ..., s^8*c, s^10, 0, 0]
        float cur = c;
        row[0] = cur;
#pragma unroll
        for (int k = 1; k <= 8; ++k) { cur *= s; row[k] = cur; }
        float s2 = s * s, s4 = s2 * s2, s8 = s4 * s4;
        row[9]  = s8 * s2;
        row[10] = 0.f;
        row[11] = 0.f;
    }
    __syncthreads();

    // --- A fragments (ISA 32-bit A 16x4 layout) ----------------------------
    // lane l: M = l%16 ; VGPR j holds K = kblock + 2*(l/16) + j
    const int m  = lane & 15;
    const int kh = (lane >> 4) * 2;     // 0 or 2
    v2f a0, a1, a2;
    a0.x = polar[wave][m][0 + kh]; a0.y = polar[wave][m][0 + kh + 1];
    a1.x = polar[wave][m][4 + kh]; a1.y = polar[wave][m][4 + kh + 1];
    a2.x = polar[wave][m][8 + kh]; a2.y = polar[wave][m][8 + kh + 1];

    const int rowBase = nBase + (lane >> 4) * 8;   // C layout: M = r + 8*(l/16)

    // --- sweep the 64 column tiles ----------------------------------------
    for (int et = 0; et < 64; ++et) {
        const int col = et * 16 + m;               // C layout: N = l%16

        // B fragments (mirror of A layout): lane l holds N=l%16,
        // VGPR j holds K = kblock + 2*(l/16) + j, read from Bmat[k][col].
        v2f b0, b1, b2;
        b0.x = Bmat[(0 + kh) * EMB + col]; b0.y = Bmat[(1 + kh) * EMB + col];
        b1.x = Bmat[(4 + kh) * EMB + col]; b1.y = Bmat[(5 + kh) * EMB + col];
        b2.x = Bmat[(8 + kh) * EMB + col]; b2.y = Bmat[(9 + kh) * EMB + col];

        v8f acc = {};
        acc = __builtin_amdgcn_wmma_f32_16x16x4_f32(false, a0, false, b0,
                                                    (short)0, acc, false, false);
        acc = __builtin_amdgcn_wmma_f32_16x16x4_f32(false, a1, false, b1,
                                                    (short)0, acc, false, false);
        acc = __builtin_amdgcn_wmma_f32_16x16x4_f32(false, a2, false, b2,
                                                    (short)0, acc, false, false);

        const float bb = bias[col];
        float* op = out + (size_t)rowBase * EMB + col;
#pragma unroll
        for (int r = 0; r < 8; ++r)
            op[(size_t)r * EMB] = acc[r] + bb;
    }
}

// ---------------------------------------------------------------------------
extern "C" void kernel_launch(void* const* d_in, const int* in_sizes, int n_in,
                              void* d_out, int out_size, void* d_ws, size_t ws_size,
                              hipStream_t stream)
{
    const float* vals = (const float*)d_in[0];   // batch_val [N]
    const float* Q    = (const float*)d_in[1];   // [10,10]
    const float* W    = (const float*)d_in[2];   // [1024,10]
    const float* b    = (const float*)d_in[3];   // [1024]
    float*       out  = (float*)d_out;           // [N,1024]
    float*       Bmat = (float*)d_ws;            // [12,1024] scratch

    const int n = in_sizes[0];

    build_B_kernel<<<(EMB + 255) / 256, 256, 0, stream>>>(W, Q, Bmat);

    const int rowsPerBlock = 8 * 16;             // 8 waves x 16 rows
    const int blocks = (n + rowsPerBlock - 1) / rowsPerBlock;
    dice_embed_kernel<<<blocks, 256, 0, stream>>>(vals, Bmat, b, out, n);
}